// ToneStack_7000796692944
// MI455X (gfx1250) — compile-verified
//
#include <hip/hip_runtime.h>
#include <math.h>

// ToneStack: 3 cascaded time-varying RBJ biquads, parallelized as a chunked
// affine scan. Phase 2 combines chunk maps with V_WMMA_F32_16X16X4_F32.

#define B_ 16
#define N_ 96000
#define L_ 250
#define C_ (N_ / L_)   // 384 chunks per batch
static_assert(N_ % L_ == 0, "chunking must be exact");

typedef __attribute__((ext_vector_type(2))) float v2f;
typedef __attribute__((ext_vector_type(8))) float v8f;

// ---- constants (double-precision evaluated, fp32 literals) ----
// low shelf: fc=120, Q=0.707, w0 = pi/200
#define CW_L   0.9998766325f     // cos(w0_low)
#define AL_L   0.011108428f      // sin(w0_low)/(2*0.707)
// high shelf: fc=4000, Q=0.707, w0 = pi/6
#define CW_H   0.8660254038f     // cos(pi/6)
#define AL_H   0.35360679f       // 0.5/(2*0.707)
#define K40    0.08304820237f    // log2(10)/40
#define W0K    1.3089969389957472e-4f  // 2*pi/48000

struct Biq { float b0, b1, b2, a1, a2; };

__device__ __forceinline__ Biq coef_low(float g) {
  float A   = exp2f(g * K40);
  float sA  = sqrtf(A);
  float t   = 2.0f * sA * AL_L;
  float ap1 = A + 1.0f, am1 = A - 1.0f;
  float am1c = am1 * CW_L, ap1c = ap1 * CW_L;
  float ra0 = 1.0f / (ap1 + am1c + t);
  Biq c;
  c.b0 = A * (ap1 - am1c + t) * ra0;
  c.b1 = 2.0f * A * (am1 - ap1c) * ra0;
  c.b2 = A * (ap1 - am1c - t) * ra0;
  c.a1 = -2.0f * (am1 + ap1c) * ra0;
  c.a2 = (ap1 + am1c - t) * ra0;
  return c;
}

__device__ __forceinline__ Biq coef_high(float g) {
  float A   = exp2f(g * K40);
  float sA  = sqrtf(A);
  float t   = 2.0f * sA * AL_H;
  float ap1 = A + 1.0f, am1 = A - 1.0f;
  float am1c = am1 * CW_H, ap1c = ap1 * CW_H;
  float ra0 = 1.0f / (ap1 - am1c + t);
  Biq c;
  c.b0 = A * (ap1 + am1c + t) * ra0;
  c.b1 = -2.0f * A * (am1 + ap1c) * ra0;
  c.b2 = A * (ap1 + am1c - t) * ra0;
  c.a1 = 2.0f * (am1 - ap1c) * ra0;
  c.a2 = (ap1 - am1c - t) * ra0;
  return c;
}

__device__ __forceinline__ Biq coef_peak(float g, float fc, float Q) {
  fc = fminf(fmaxf(fc, 50.0f), 8000.0f);
  Q  = fminf(fmaxf(Q, 0.2f), 4.0f);
  float w0 = fc * W0K;
  float sw = sinf(w0), cw = cosf(w0);
  float A     = exp2f(g * K40);
  float alpha = sw / (2.0f * Q);
  float aA  = alpha * A;
  float adA = alpha / A;
  float ra0 = 1.0f / (1.0f + adA);
  Biq c;
  c.b0 = (1.0f + aA) * ra0;
  c.b1 = -2.0f * cw * ra0;
  c.b2 = (1.0f - aA) * ra0;
  c.a1 = c.b1;               // a1 == b1 for RBJ peaking (both -2cos/a0)
  c.a2 = (1.0f - adA) * ra0;
  return c;
}

// ---------------------------------------------------------------------------
// Phase 1: per-chunk serial composition of per-sample affine maps.
// State s = (z1l,z2l,z1m,z2m,z1h,z2h);  s' = A_n s + x_n g_n.
// M stored column-major as 6x7 (A | v).  Record = 6 rows x 8 floats.
// ---------------------------------------------------------------------------
__global__ void __launch_bounds__(256) ts_maps(
    const float* __restrict__ x,  const float* __restrict__ gl,
    const float* __restrict__ gm, const float* __restrict__ fc,
    const float* __restrict__ qq, const float* __restrict__ gh,
    float* __restrict__ maps)
{
  int t = blockIdx.x * blockDim.x + threadIdx.x;
  if (t >= B_ * C_) return;
  int b = t / C_, ci = t % C_;
  const size_t base = (size_t)b * N_ + (size_t)ci * L_;

  float M[7][6];  // M[col][row]
#pragma unroll
  for (int c = 0; c < 7; ++c)
#pragma unroll
    for (int r = 0; r < 6; ++r)
      M[c][r] = (c == r) ? 1.0f : 0.0f;

  for (int i = 0; i < L_; ++i) {
    size_t idx = base + i;
    float xi = x[idx];
    Biq l = coef_low(gl[idx]);
    Biq m = coef_peak(gm[idx], fc[idx], qq[idx]);
    Biq h = coef_high(gh[idx]);
    float p1l = fmaf(-l.a1, l.b0, l.b1), p2l = fmaf(-l.a2, l.b0, l.b2);
    float p1m = fmaf(-m.a1, m.b0, m.b1), p2m = fmaf(-m.a2, m.b0, m.b2);
    float p1h = fmaf(-h.a1, h.b0, h.b1), p2h = fmaf(-h.a2, h.b0, h.b2);

#pragma unroll
    for (int c = 0; c < 7; ++c) {
      float u = (c == 6) ? xi : 0.0f;  // input enters only the translation col
      float w0 = M[c][0], w1 = M[c][1], w2 = M[c][2];
      float w3 = M[c][3], w4 = M[c][4], w5 = M[c][5];
      float vm = fmaf(l.b0, u, w0);      // column coeff of u_mid
      float vh = fmaf(m.b0, vm, w2);     // column coeff of u_high
      M[c][0] = fmaf(p1l, u, fmaf(-l.a1, w0, w1));
      M[c][1] = fmaf(p2l, u, -l.a2 * w0);
      M[c][2] = fmaf(p1m, vm, fmaf(-m.a1, w2, w3));
      M[c][3] = fmaf(p2m, vm, -m.a2 * w2);
      M[c][4] = fmaf(p1h, vh, fmaf(-h.a1, w4, w5));
      M[c][5] = fmaf(p2h, vh, -h.a2 * w4);
    }
  }

  float* rec = maps + (size_t)t * 48;  // 6 rows x 8 floats, row-major
#pragma unroll
  for (int r = 0; r < 6; ++r) {
#pragma unroll
    for (int c = 0; c < 7; ++c) rec[r * 8 + c] = M[c][r];
    rec[r * 8 + 7] = 0.0f;
  }
}

// ---------------------------------------------------------------------------
// Phase 2: per-batch serial scan over chunk maps with WMMA composition.
// P (running 7x7 homogeneous map, padded into 16x16) lives in LDS.
// P' = T_c * P via two chained V_WMMA_F32_16X16X4_F32 (K blocks 0..3, 4..7).
// Exclusive prefix (state at chunk start) = column 6 of P (initial state = 0).
// One wave (32 threads) per batch; in-wave LDS ordering -> no barriers.
// ---------------------------------------------------------------------------
__global__ void __launch_bounds__(32) ts_scan(const float* __restrict__ maps,
                                              float* __restrict__ starts)
{
  __shared__ float P[16 * 16];
  const int lane = threadIdx.x;
  const int b    = blockIdx.x;
  const int m    = lane & 15;        // A row / B col handled by this lane
  const int half = lane >> 4;
  const int koff = half * 2;         // K offset within a K=4 block
  const int n    = m;

#pragma unroll
  for (int k = 0; k < 8; ++k) {
    int e = lane * 8 + k;
    int r = e >> 4, cc = e & 15;
    P[e] = (r == cc && r < 7) ? 1.0f : 0.0f;  // 7x7 identity
  }

  for (int c = 0; c < C_; ++c) {
    const float* rec = maps + (size_t)(b * C_ + c) * 48;

    // exclusive prefix: state entering chunk c = translation column of P
    if (lane < 6)
      starts[(size_t)(b * C_ + c) * 6 + lane] = P[lane * 16 + 6];

    // A operand = this chunk's map T (7x7 padded; row 6 = e6, rows 7..15 = 0)
    float a00, a01, a10, a11;
    if (m < 6) {
      a00 = rec[m * 8 + koff];
      a01 = rec[m * 8 + koff + 1];
      a10 = rec[m * 8 + 4 + koff];
      a11 = rec[m * 8 + 4 + koff + 1];
    } else if (m == 6) {
      a00 = 0.0f; a01 = 0.0f;
      a10 = (koff == 2) ? 1.0f : 0.0f;  // T[6][6] = 1 (block1, col 6)
      a11 = 0.0f;
    } else {
      a00 = a01 = a10 = a11 = 0.0f;
    }

    // B operand = running P from LDS (rows koff, koff+1 per K block)
    float b00 = P[(koff + 0) * 16 + n];
    float b01 = P[(koff + 1) * 16 + n];
    float b10 = P[(koff + 4) * 16 + n];
    float b11 = P[(koff + 5) * 16 + n];

    v2f A0 = {a00, a01}, A1 = {a10, a11};
    v2f B0 = {b00, b01}, B1 = {b10, b11};
    v8f acc = {0.f, 0.f, 0.f, 0.f, 0.f, 0.f, 0.f, 0.f};
    acc = __builtin_amdgcn_wmma_f32_16x16x4_f32(false, A0, false, B0,
                                                (short)0, acc, false, false);
    acc = __builtin_amdgcn_wmma_f32_16x16x4_f32(false, A1, false, B1,
                                                (short)0, acc, false, false);

    // D back to LDS in C/D layout: vgpr v -> row v + 8*half, col n
#pragma unroll
    for (int v = 0; v < 8; ++v)
      P[(v + 8 * half) * 16 + n] = acc[v];
  }
}

// ---------------------------------------------------------------------------
// Phase 3: re-run each chunk serially from its start state, emit output.
// Inputs were just streamed in phase 1 (37 MB) -> re-reads hit the 192 MB L2.
// ---------------------------------------------------------------------------
__global__ void __launch_bounds__(256) ts_emit(
    const float* __restrict__ x,  const float* __restrict__ gl,
    const float* __restrict__ gm, const float* __restrict__ fc,
    const float* __restrict__ qq, const float* __restrict__ gh,
    const float* __restrict__ starts, float* __restrict__ out)
{
  int t = blockIdx.x * blockDim.x + threadIdx.x;
  if (t >= B_ * C_) return;
  int b = t / C_, ci = t % C_;
  const float* s = starts + (size_t)t * 6;
  float z1l = s[0], z2l = s[1], z1m = s[2], z2m = s[3], z1h = s[4], z2h = s[5];
  const size_t base = (size_t)b * N_ + (size_t)ci * L_;

  for (int i = 0; i < L_; ++i) {
    size_t idx = base + i;
    float xi = x[idx];
    Biq l = coef_low(gl[idx]);
    Biq m = coef_peak(gm[idx], fc[idx], qq[idx]);
    Biq h = coef_high(gh[idx]);

    float yl   = fmaf(l.b0, xi, z1l);
    float nz1l = fmaf(l.b1, xi, fmaf(-l.a1, yl, z2l));
    float nz2l = fmaf(l.b2, xi, -l.a2 * yl);
    float ym   = fmaf(m.b0, yl, z1m);
    float nz1m = fmaf(m.b1, yl, fmaf(-m.a1, ym, z2m));
    float nz2m = fmaf(m.b2, yl, -m.a2 * ym);
    float yh   = fmaf(h.b0, ym, z1h);
    float nz1h = fmaf(h.b1, ym, fmaf(-h.a1, yh, z2h));
    float nz2h = fmaf(h.b2, ym, -h.a2 * yh);
    z1l = nz1l; z2l = nz2l; z1m = nz1m; z2m = nz2m; z1h = nz1h; z2h = nz2h;

    out[idx] = yh;
  }
}

extern "C" void kernel_launch(void* const* d_in, const int* in_sizes, int n_in,
                              void* d_out, int out_size, void* d_ws, size_t ws_size,
                              hipStream_t stream)
{
  const float* x  = (const float*)d_in[0];
  const float* gl = (const float*)d_in[1];
  const float* gm = (const float*)d_in[2];
  const float* fc = (const float*)d_in[3];
  const float* qq = (const float*)d_in[4];
  const float* gh = (const float*)d_in[5];
  float* out = (float*)d_out;

  const size_t starts_floats = (size_t)B_ * C_ * 6;   //   144 KB
  const size_t maps_floats   = (size_t)B_ * C_ * 48;  //  1.18 MB
  float* starts = (float*)d_ws;
  float* maps;
  if (ws_size >= (starts_floats + maps_floats) * sizeof(float)) {
    maps = starts + starts_floats;
  } else {
    // Stage maps in d_out (6.1 MB): consumed by ts_scan before ts_emit
    // overwrites it. Deterministic host-side choice (ws_size is constant).
    maps = out;
  }

  const int work = B_ * C_;
  const int blk  = 256;
  const int grd  = (work + blk - 1) / blk;

  ts_maps<<<grd, blk, 0, stream>>>(x, gl, gm, fc, qq, gh, maps);
  ts_scan<<<B_, 32, 0, stream>>>(maps, starts);
  ts_emit<<<grd, blk, 0, stream>>>(x, gl, gm, fc, qq, gh, starts, out);
}